// SimplEx_14224931684810
// MI455X (gfx1250) — compile-verified
//
#include <hip/hip_runtime.h>
#include <math.h>

typedef __attribute__((ext_vector_type(2))) float v2f;
typedef __attribute__((ext_vector_type(8))) float v8f;

#define DIM   2048
#define NTEST 128

// ---------------------------------------------------------------- init
__global__ __launch_bounds__(256)
void simplex_init(float* __restrict__ P, float* __restrict__ M,
                  float* __restrict__ V, float* __restrict__ W) {
  int i = blockIdx.x * 256 + threadIdx.x;
  P[i] = 0.0f; M[i] = 0.0f; V[i] = 0.0f;
  W[i] = 1.0f / 2048.0f;                 // softmax(0) row-uniform
}

// ---------------------------------------------------------------- transpose
// HtT[n*DIM + k] = Ht[k*DIM + n]  (32x32 LDS tiles, coalesced both ways)
__global__ __launch_bounds__(256)
void simplex_transpose(const float* __restrict__ src, float* __restrict__ dst) {
  __shared__ float tile[32][33];
  const int tx = threadIdx.x & 31;
  const int ty = threadIdx.x >> 5;           // 8 rows per pass
  const int bx = blockIdx.x * 32;            // src col base
  const int by = blockIdx.y * 32;            // src row base
#pragma unroll
  for (int r = 0; r < 32; r += 8)
    tile[ty + r][tx] = src[(by + ty + r) * DIM + bx + tx];
  __syncthreads();
#pragma unroll
  for (int r = 0; r < 32; r += 8)
    dst[(bx + ty + r) * DIM + by + tx] = tile[tx][ty + r];
}

// ---------------------------------------------------------------- forward (transposed-B fast path)
// R = W @ Ht - H, with BT = Ht^T so B[K,N] = BT[N*DIM+K] -> contiguous b64 loads
__global__ __launch_bounds__(128)
void simplex_fwd_t(const float* __restrict__ W, const float* __restrict__ BT,
                   const float* __restrict__ H, float* __restrict__ R) {
  const int lane = threadIdx.x & 31;
  const int wave = threadIdx.x >> 5;
  const int r16  = lane & 15;     // M for A-frag / N for B-frag / N for D
  const int khi  = lane >> 4;     // lanes 16-31 carry K=2,3 of each K-step
  const int Mb   = blockIdx.y * 32;
  const int Nb   = blockIdx.x * 128 + wave * 32;

  v8f c00 = {}; v8f c01 = {}; v8f c10 = {}; v8f c11 = {};

  const int aoff0 = (Mb + r16) * DIM + 2 * khi;
  const int aoff1 = aoff0 + 16 * DIM;
  const int boff0 = (Nb + r16) * DIM + 2 * khi;
  const int boff1 = boff0 + 16 * DIM;

#pragma unroll 4
  for (int k = 0; k < DIM; k += 4) {
    v2f a0 = *(const v2f*)(W  + aoff0 + k);
    v2f a1 = *(const v2f*)(W  + aoff1 + k);
    v2f b0 = *(const v2f*)(BT + boff0 + k);
    v2f b1 = *(const v2f*)(BT + boff1 + k);
    c00 = __builtin_amdgcn_wmma_f32_16x16x4_f32(false, a0, false, b0, (short)0, c00, false, false);
    c01 = __builtin_amdgcn_wmma_f32_16x16x4_f32(false, a0, false, b1, (short)0, c01, false, false);
    c10 = __builtin_amdgcn_wmma_f32_16x16x4_f32(false, a1, false, b0, (short)0, c10, false, false);
    c11 = __builtin_amdgcn_wmma_f32_16x16x4_f32(false, a1, false, b1, (short)0, c11, false, false);
  }

  const int col0 = Nb + r16;
  const int mhi  = khi * 8;                 // D: lanes 16-31 hold M = v+8
#pragma unroll
  for (int v = 0; v < 8; ++v) {
    const int row0 = Mb + mhi + v;
    const int row1 = row0 + 16;
    R[row0 * DIM + col0]      = c00[v] - H[row0 * DIM + col0];
    R[row0 * DIM + col0 + 16] = c01[v] - H[row0 * DIM + col0 + 16];
    R[row1 * DIM + col0]      = c10[v] - H[row1 * DIM + col0];
    R[row1 * DIM + col0 + 16] = c11[v] - H[row1 * DIM + col0 + 16];
  }
}

// ---------------------------------------------------------------- forward (fallback, strided B)
__global__ __launch_bounds__(128)
void simplex_fwd(const float* __restrict__ W, const float* __restrict__ Ht,
                 const float* __restrict__ H, float* __restrict__ R) {
  const int lane = threadIdx.x & 31;
  const int wave = threadIdx.x >> 5;
  const int r16  = lane & 15;
  const int khi  = lane >> 4;
  const int Mb   = blockIdx.y * 32;
  const int Nb   = blockIdx.x * 128 + wave * 32;

  v8f c00 = {}; v8f c01 = {}; v8f c10 = {}; v8f c11 = {};

  const int aoff0 = (Mb + r16) * DIM + 2 * khi;
  const int aoff1 = aoff0 + 16 * DIM;
  const int boff  = (2 * khi) * DIM + (Nb + r16);

#pragma unroll 4
  for (int k = 0; k < DIM; k += 4) {
    v2f a0 = *(const v2f*)(W + aoff0 + k);
    v2f a1 = *(const v2f*)(W + aoff1 + k);
    const int bk = boff + k * DIM;
    v2f b0, b1;
    b0.x = Ht[bk];            b0.y = Ht[bk + DIM];
    b1.x = Ht[bk + 16];       b1.y = Ht[bk + DIM + 16];
    c00 = __builtin_amdgcn_wmma_f32_16x16x4_f32(false, a0, false, b0, (short)0, c00, false, false);
    c01 = __builtin_amdgcn_wmma_f32_16x16x4_f32(false, a0, false, b1, (short)0, c01, false, false);
    c10 = __builtin_amdgcn_wmma_f32_16x16x4_f32(false, a1, false, b0, (short)0, c10, false, false);
    c11 = __builtin_amdgcn_wmma_f32_16x16x4_f32(false, a1, false, b1, (short)0, c11, false, false);
  }

  const int col0 = Nb + r16;
  const int mhi  = khi * 8;
#pragma unroll
  for (int v = 0; v < 8; ++v) {
    const int row0 = Mb + mhi + v;
    const int row1 = row0 + 16;
    R[row0 * DIM + col0]      = c00[v] - H[row0 * DIM + col0];
    R[row0 * DIM + col0 + 16] = c01[v] - H[row0 * DIM + col0 + 16];
    R[row1 * DIM + col0]      = c10[v] - H[row1 * DIM + col0];
    R[row1 * DIM + col0 + 16] = c11[v] - H[row1 * DIM + col0 + 16];
  }
}

// ---------------------------------------------------------------- backward
// G = 2 * R @ Ht^T   -> B[K,N] = Ht[N*DIM + K]  (contiguous b64 B loads)
__global__ __launch_bounds__(128)
void simplex_bwd(const float* __restrict__ Rm, const float* __restrict__ Ht,
                 float* __restrict__ G) {
  const int lane = threadIdx.x & 31;
  const int wave = threadIdx.x >> 5;
  const int r16  = lane & 15;
  const int khi  = lane >> 4;
  const int Mb   = blockIdx.y * 32;
  const int Nb   = blockIdx.x * 128 + wave * 32;

  v8f c00 = {}; v8f c01 = {}; v8f c10 = {}; v8f c11 = {};

  const int aoff0 = (Mb + r16) * DIM + 2 * khi;
  const int aoff1 = aoff0 + 16 * DIM;
  const int boff0 = (Nb + r16) * DIM + 2 * khi;
  const int boff1 = boff0 + 16 * DIM;

#pragma unroll 4
  for (int k = 0; k < DIM; k += 4) {
    v2f a0 = *(const v2f*)(Rm + aoff0 + k);
    v2f a1 = *(const v2f*)(Rm + aoff1 + k);
    v2f b0 = *(const v2f*)(Ht + boff0 + k);
    v2f b1 = *(const v2f*)(Ht + boff1 + k);
    c00 = __builtin_amdgcn_wmma_f32_16x16x4_f32(false, a0, false, b0, (short)0, c00, false, false);
    c01 = __builtin_amdgcn_wmma_f32_16x16x4_f32(false, a0, false, b1, (short)0, c01, false, false);
    c10 = __builtin_amdgcn_wmma_f32_16x16x4_f32(false, a1, false, b0, (short)0, c10, false, false);
    c11 = __builtin_amdgcn_wmma_f32_16x16x4_f32(false, a1, false, b1, (short)0, c11, false, false);
  }

  const int col0 = Nb + r16;
  const int mhi  = khi * 8;
#pragma unroll
  for (int v = 0; v < 8; ++v) {
    const int row0 = Mb + mhi + v;
    const int row1 = row0 + 16;
    G[row0 * DIM + col0]      = 2.0f * c00[v];
    G[row0 * DIM + col0 + 16] = 2.0f * c01[v];
    G[row1 * DIM + col0]      = 2.0f * c10[v];
    G[row1 * DIM + col0 + 16] = 2.0f * c11[v];
  }
}

// ---------------------------------------------------------------- update
// Per row: s = <W,G>; g = W*(G - s); Adam(P,m,v); W = softmax(P_new)
__global__ __launch_bounds__(256)
void simplex_upd(float* __restrict__ P, float* __restrict__ Madam,
                 float* __restrict__ Vadam, float* __restrict__ W,
                 const float* __restrict__ G, float bc1, float bc2) {
  __shared__ float red[256];
  const int row = blockIdx.x;
  const int tid = threadIdx.x;
  const int base = row * DIM;

  float w[8], g[8];
  float s = 0.0f;
#pragma unroll
  for (int e = 0; e < 8; ++e) {
    const int j = tid + e * 256;
    w[e] = W[base + j];
    g[e] = G[base + j];
    s += w[e] * g[e];
  }
  red[tid] = s; __syncthreads();
  for (int off = 128; off > 0; off >>= 1) {
    if (tid < off) red[tid] += red[tid + off];
    __syncthreads();
  }
  s = red[0]; __syncthreads();

  float pnew[8];
  float mx = -INFINITY;
#pragma unroll
  for (int e = 0; e < 8; ++e) {
    const int idx = base + tid + e * 256;
    const float grad = w[e] * (g[e] - s);
    const float mm = 0.9f   * Madam[idx] + 0.1f   * grad;
    const float vv = 0.999f * Vadam[idx] + 0.001f * grad * grad;
    Madam[idx] = mm; Vadam[idx] = vv;
    const float p = P[idx] - 1e-3f * (mm * bc1) / (sqrtf(vv * bc2) + 1e-8f);
    P[idx] = p; pnew[e] = p;
    mx = fmaxf(mx, p);
  }
  red[tid] = mx; __syncthreads();
  for (int off = 128; off > 0; off >>= 1) {
    if (tid < off) red[tid] = fmaxf(red[tid], red[tid + off]);
    __syncthreads();
  }
  mx = red[0]; __syncthreads();

  float ex[8];
  float es = 0.0f;
#pragma unroll
  for (int e = 0; e < 8; ++e) { ex[e] = expf(pnew[e] - mx); es += ex[e]; }
  red[tid] = es; __syncthreads();
  for (int off = 128; off > 0; off >>= 1) {
    if (tid < off) red[tid] += red[tid + off];
    __syncthreads();
  }
  es = red[0];
  const float inv = 1.0f / es;
#pragma unroll
  for (int e = 0; e < 8; ++e) W[base + tid + e * 256] = ex[e] * inv;
}

// ---------------------------------------------------------------- launch
extern "C" void kernel_launch(void* const* d_in, const int* in_sizes, int n_in,
                              void* d_out, int out_size, void* d_ws, size_t ws_size,
                              hipStream_t stream) {
  const float* H  = (const float*)d_in[0];   // [128, 2048]
  const float* Ht = (const float*)d_in[1];   // [2048, 2048]
  float* Wout = (float*)d_out;               // [128, 2048] = softmax(P) (live state)

  const size_t n = (size_t)NTEST * DIM;      // 262144
  float* P = (float*)d_ws;
  float* M = P + n;
  float* V = M + n;
  float* R = V + n;
  float* G = R + n;
  float* HtT = G + n;                        // optional 16 MB transposed copy

  const bool haveT =
      ws_size >= 5u * n * sizeof(float) + (size_t)DIM * DIM * sizeof(float);

  simplex_init<<<dim3((unsigned)(n / 256)), 256, 0, stream>>>(P, M, V, Wout);
  if (haveT)
    simplex_transpose<<<dim3(64, 64), 256, 0, stream>>>(Ht, HtT);

  const dim3 ggrid(16, 4);   // N-superblocks of 128 cols x M-blocks of 32 rows
  const dim3 gblk(128);

  double b1t = 1.0, b2t = 1.0;
  for (int t = 1; t <= 1000; ++t) {
    b1t *= 0.9; b2t *= 0.999;
    const float bc1 = (float)(1.0 / (1.0 - b1t));
    const float bc2 = (float)(1.0 / (1.0 - b2t));
    if (haveT)
      simplex_fwd_t<<<ggrid, gblk, 0, stream>>>(Wout, HtT, H, R);
    else
      simplex_fwd<<<ggrid, gblk, 0, stream>>>(Wout, Ht, H, R);
    simplex_bwd<<<ggrid, gblk, 0, stream>>>(R, Ht, G);
    simplex_upd<<<NTEST, 256, 0, stream>>>(P, M, V, Wout, G, bc1, bc2);
  }
}